// Model_33002528702883
// MI455X (gfx1250) — compile-verified
//
#include <hip/hip_runtime.h>

#define BLOCK 256
#define GRID 2048
#define WAVES (BLOCK / 32)
#define MAX_LDS_BINS 1024
#define REPLICAS 4
#define CHUNK 4                        // async b128 issues per lane per tile
#define TILE_V4 (32 * CHUNK)           // float4s per wave-tile (128)
#define TILE_FLOATS (TILE_V4 * 4)      // 512 floats per wave-tile

// ---- order-preserving IEEE-754 <-> uint encode (unsigned compare == float compare) ----
__device__ __forceinline__ unsigned enc_f32(float f) {
    unsigned u = __float_as_uint(f);
    return (u & 0x80000000u) ? ~u : (u | 0x80000000u);
}
__device__ __forceinline__ float dec_f32(unsigned e) {
    unsigned u = (e & 0x80000000u) ? (e ^ 0x80000000u) : ~e;
    return __uint_as_float(u);
}

// ---- pass 0: reset workspace (min sentinel, max sentinel, zero bins) ----
__global__ void hist_init(unsigned* __restrict__ ws, unsigned ws_words) {
    unsigned i = blockIdx.x * blockDim.x + threadIdx.x;
    unsigned want = 2u + MAX_LDS_BINS;
    unsigned lim = ws_words < want ? ws_words : want;
    if (i == 0 && ws_words > 0) ws[0] = 0xFFFFFFFFu;   // running-min sentinel (encoded)
    if (i == 1 && ws_words > 1) ws[1] = 0u;            // running-max sentinel (encoded)
    for (unsigned k = i; k < lim; k += gridDim.x * blockDim.x)
        if (k >= 2) ws[k] = 0u;                        // bin counters
}

// ---- pass 1: global min/max via b128 streaming + wave32 shuffle reduce ----
__global__ void minmax_kernel(const float* __restrict__ x, long long n,
                              unsigned* __restrict__ ws) {
    unsigned lmin = 0xFFFFFFFFu, lmax = 0u;
    long long tid    = (long long)blockIdx.x * blockDim.x + threadIdx.x;
    long long stride = (long long)gridDim.x * blockDim.x;
    long long n4 = n >> 2;
    const float4* __restrict__ x4 = (const float4*)x;

    for (long long i = tid; i < n4; i += stride) {
        if (i + 8 * stride < n4)
            __builtin_prefetch((const void*)&x4[i + 8 * stride], 0, 0);  // global_prefetch_b8
        float4 v = x4[i];                                                // global_load_b128
        unsigned e0 = enc_f32(v.x), e1 = enc_f32(v.y);
        unsigned e2 = enc_f32(v.z), e3 = enc_f32(v.w);
        unsigned a = min(e0, e1), b = min(e2, e3);
        unsigned c = max(e0, e1), d = max(e2, e3);
        lmin = min(lmin, min(a, b));
        lmax = max(lmax, max(c, d));
    }
    for (long long i = (n4 << 2) + tid; i < n; i += stride) {
        unsigned e = enc_f32(x[i]);
        lmin = min(lmin, e);
        lmax = max(lmax, e);
    }

    // wave32 butterfly reduction (CDNA5 is wave32-only)
    #pragma unroll
    for (int off = 16; off > 0; off >>= 1) {
        lmin = min(lmin, (unsigned)__shfl_xor((int)lmin, off, 32));
        lmax = max(lmax, (unsigned)__shfl_xor((int)lmax, off, 32));
    }

    __shared__ unsigned smin[WAVES], smax[WAVES];
    int lane = threadIdx.x & 31, wid = threadIdx.x >> 5;
    if (lane == 0) { smin[wid] = lmin; smax[wid] = lmax; }
    __syncthreads();
    if (threadIdx.x == 0) {
        unsigned m = smin[0], M = smax[0];
        #pragma unroll
        for (int i = 1; i < WAVES; ++i) { m = min(m, smin[i]); M = max(M, smax[i]); }
        atomicMin(&ws[0], m);   // global_atomic_min_u32
        atomicMax(&ws[1], M);   // global_atomic_max_u32
    }
}

// ---- pass 2: async global->LDS double-buffered staging + replicated LDS histograms ----
__global__ void hist_kernel(const float* __restrict__ x, long long n,
                            const int* __restrict__ binsp,
                            unsigned* __restrict__ ws) {
    __shared__ unsigned sh_hist[REPLICAS * MAX_LDS_BINS];
    __shared__ float4   sh_stage[2][WAVES][TILE_V4];

    int bins = *binsp;
    if (bins < 1) bins = 1;

    float hmin = dec_f32(ws[0]);
    float hmax = dec_f32(ws[1]);
    if (hmax == hmin) hmax = hmin + 1.0f;   // degenerate-range guard (matches reference)
    float scale = (float)bins / (hmax - hmin);

    long long n4 = n >> 2;
    int lane = threadIdx.x & 31, wid = threadIdx.x >> 5;

    if (bins <= MAX_LDS_BINS) {
        for (int i = threadIdx.x; i < REPLICAS * bins; i += blockDim.x) sh_hist[i] = 0u;
        __syncthreads();
        unsigned rep = ((unsigned)wid & (REPLICAS - 1)) * (unsigned)bins;

        // wave-relative LDS byte offsets of this wave's two staging buffers
        unsigned ldsOff[2];
        ldsOff[0] = (unsigned)(uintptr_t)(&sh_stage[0][wid][0]);
        ldsOff[1] = (unsigned)(uintptr_t)(&sh_stage[1][wid][0]);

        long long numTiles  = (n4 + TILE_V4 - 1) / TILE_V4;
        long long waveId    = (long long)blockIdx.x * WAVES + wid;
        long long waveCount = (long long)gridDim.x * WAVES;

        // helper lambda: issue one tile's CHUNK async b128 copies into buffer `buf`
        auto issue_tile = [&](long long tile, int buf) {
            long long v4base = tile * (long long)TILE_V4;
            #pragma unroll
            for (int k = 0; k < CHUNK; ++k) {
                long long idx = v4base + (long long)(k * 32 + lane);
                if (idx >= n4) idx = n4 - 1;            // clamp: keep EXEC full, counts exact
                unsigned goff = (unsigned)(idx * 16);   // byte offset from base (n*4 < 2^31)
                unsigned loff = ldsOff[buf] + (unsigned)(k * 32 + lane) * 16u;
                asm volatile("global_load_async_to_lds_b128 %0, %1, %2 offset:0"
                             :: "v"(loff), "v"(goff), "s"(x)
                             : "memory");               // ASYNCcnt += 1
            }
        };

        auto process_tile = [&](long long tile, int buf) {
            int validV4 = (int)((n4 - tile * (long long)TILE_V4) < TILE_V4
                                    ? (n4 - tile * (long long)TILE_V4) : TILE_V4);
            const float4* stage = &sh_stage[buf][wid][0];
            #pragma unroll
            for (int k = 0; k < CHUNK; ++k) {
                int s = k * 32 + lane;
                if (s < validV4) {
                    float4 v = stage[s];                // ds_load_b128
                    int b0 = (int)floorf((v.x - hmin) * scale);
                    int b1 = (int)floorf((v.y - hmin) * scale);
                    int b2 = (int)floorf((v.z - hmin) * scale);
                    int b3 = (int)floorf((v.w - hmin) * scale);
                    b0 = min(max(b0, 0), bins - 1);
                    b1 = min(max(b1, 0), bins - 1);
                    b2 = min(max(b2, 0), bins - 1);
                    b3 = min(max(b3, 0), bins - 1);
                    atomicAdd(&sh_hist[rep + b0], 1u);  // ds_add_u32
                    atomicAdd(&sh_hist[rep + b1], 1u);
                    atomicAdd(&sh_hist[rep + b2], 1u);
                    atomicAdd(&sh_hist[rep + b3], 1u);
                }
            }
        };

        if (waveId < numTiles) {
            issue_tile(waveId, 0);                      // prologue: fill buffer 0
            int cur = 0;
            for (long long t = waveId; t < numTiles; t += waveCount) {
                long long tn = t + waveCount;
                bool hasNext = tn < numTiles;
                if (hasNext) issue_tile(tn, cur ^ 1);   // keep next tile's DMA in flight
                if (hasNext)
                    asm volatile("s_wait_asynccnt 0x4" ::: "memory"); // cur tile landed
                else
                    asm volatile("s_wait_asynccnt 0x0" ::: "memory"); // drain
                process_tile(t, cur);
                cur ^= 1;
            }
        }

        // tail elements (n % 4): direct global atomics from block 0
        if (blockIdx.x == 0 && threadIdx.x < (int)(n & 3)) {
            int b = (int)floorf((x[(n4 << 2) + threadIdx.x] - hmin) * scale);
            b = min(max(b, 0), bins - 1);
            atomicAdd(&ws[2 + b], 1u);
        }

        __syncthreads();
        // fold replicas, one global atomic per touched bin per block
        for (int i = threadIdx.x; i < bins; i += blockDim.x) {
            unsigned s = 0;
            #pragma unroll
            for (int r = 0; r < REPLICAS; ++r) s += sh_hist[r * bins + i];
            if (s) atomicAdd(&ws[2 + i], s);            // global_atomic_add_u32
        }
    } else {
        // fallback: direct global atomics (not exercised for bins=100)
        long long tid    = (long long)blockIdx.x * blockDim.x + threadIdx.x;
        long long stride = (long long)gridDim.x * blockDim.x;
        for (long long i = tid; i < n; i += stride) {
            int b = (int)floorf((x[i] - hmin) * scale);
            b = min(max(b, 0), bins - 1);
            atomicAdd(&ws[2 + b], 1u);
        }
    }
}

// ---- pass 3: integer counts -> fp32 output ----
__global__ void hist_finalize(const unsigned* __restrict__ ws,
                              float* __restrict__ out,
                              const int* __restrict__ binsp, int out_n) {
    int bins = *binsp;
    int i = blockIdx.x * blockDim.x + threadIdx.x;
    if (i < out_n) out[i] = (i < bins) ? (float)ws[2 + i] : 0.0f;
}

extern "C" void kernel_launch(void* const* d_in, const int* in_sizes, int n_in,
                              void* d_out, int out_size, void* d_ws, size_t ws_size,
                              hipStream_t stream) {
    const float* x     = (const float*)d_in[0];
    const int*   binsp = (const int*)d_in[1];
    long long n = (long long)in_sizes[0];
    unsigned* ws = (unsigned*)d_ws;
    float* out = (float*)d_out;
    unsigned ws_words = (unsigned)(ws_size / sizeof(unsigned));

    hist_init<<<8, BLOCK, 0, stream>>>(ws, ws_words);
    minmax_kernel<<<GRID, BLOCK, 0, stream>>>(x, n, ws);
    hist_kernel<<<GRID, BLOCK, 0, stream>>>(x, n, binsp, ws);
    hist_finalize<<<(out_size + BLOCK - 1) / BLOCK, BLOCK, 0, stream>>>(ws, out, binsp, out_size);
}